// FeatureLoss_13735305412942
// MI455X (gfx1250) — compile-verified
//
#include <hip/hip_runtime.h>
#include <cstdint>

#define RAD   5
#define DIA   11
#define TILE  16
#define HALO  26      // TILE + 2*RAD
#define NCLS  21
#define HW    128
#define NBLK  128     // 8*8*2 loss workgroups

typedef __attribute__((ext_vector_type(2))) float v2f;
typedef __attribute__((ext_vector_type(8))) float v8f;

// WMMA-based wave sum: A(16x4) vgpr0 = val (K=0 lanes 0-15 / K=2 lanes
// 16-31), vgpr1 = 0, B = ones -> every column of D is the row-sum vector;
// this lane's 8 D regs + partner lane (lane^16 via ds_swizzle SWAPX16) give
// the exact f32 wave total in every lane. Requires EXEC == all ones.
__device__ __forceinline__ float wave_sum_wmma(float val) {
    v2f a; a[0] = val;  a[1] = 0.0f;
    v2f b; b[0] = 1.0f; b[1] = 1.0f;
    v8f cz = {};
    v8f d = __builtin_amdgcn_wmma_f32_16x16x4_f32(
        false, a, false, b, (short)0, cz, false, false);
    float rs = d[0] + d[1] + d[2] + d[3] + d[4] + d[5] + d[6] + d[7];
    rs += __int_as_float(
        __builtin_amdgcn_ds_swizzle(__float_as_int(rs), 0x401F)); // lane ^ 16
    return rs;
}

// ---------------------------------------------------------------------------
// Kernel 1: 4x4 average pool of rgb (2,3,512,512) -> (2,3,128,128), scaled by
// 10 = 1/sigma_rgb so the loss kernel consumes features directly.
// ---------------------------------------------------------------------------
__global__ __launch_bounds__(256) void rgb_pool_kernel(
    const float* __restrict__ rgb, float* __restrict__ out)
{
    int i = blockIdx.x * 256 + threadIdx.x;          // over 2*3*128*128
    if (i >= 2 * 3 * HW * HW) return;
    int x  = i & (HW - 1);
    int y  = (i >> 7) & (HW - 1);
    int nc = i >> 14;
    const float* base = rgb + ((size_t)nc * 512 + (size_t)y * 4) * 512 + (size_t)x * 4;
    float s = 0.f;
#pragma unroll
    for (int r = 0; r < 4; ++r)
#pragma unroll
        for (int c = 0; c < 4; ++c)
            s += base[r * 512 + c];
    out[i] = s * 0.625f;                              // (1/16) * 10
}

// ---------------------------------------------------------------------------
// Kernel 2: tiled CRF loss. One 16x16 tile / workgroup (256 thr = 8 waves).
// Halo (26x26) of y (21 ch) + rgb_ds (3 ch) staged into LDS with CDNA5 async
// loads; out-of-image halo cells are zero-filled (== reference zero padding).
// y classes stored as 5 pixel-contiguous float4 banks + 1 scalar channel so
// the inner loop uses ds_load_b128. Block partial -> d_ws (deterministic
// two-stage reduction, no float atomics).
// ---------------------------------------------------------------------------
__global__ __launch_bounds__(256, 4) void crf_loss_kernel(
    const float* __restrict__ yprob,   // (2,21,128,128)
    const float* __restrict__ rgbds,   // (2,3,128,128), pre-scaled by 10
    float* __restrict__ partials)      // NBLK entries
{
    __shared__ __attribute__((aligned(16))) float sy4[5][HALO][HALO][4]; // 54080 B
    __shared__ float sy1[HALO][HALO];                                    //  2704 B
    __shared__ float sr[3][HALO][HALO];                                  //  8112 B
    __shared__ float sgx[DIA + 1];                                       //    48 B
    __shared__ float swsum[8];

    const int tid = threadIdx.x;
    const int tx = blockIdx.x, ty = blockIdx.y, n = blockIdx.z;
    const int gx0 = tx * TILE - RAD;
    const int gy0 = ty * TILE - RAD;

    // dx gaussian table exp(-(dx-5)^2/72) -> LDS (broadcast reads later)
    if (tid < DIA)
        sgx[tid] = __expf((float)(-(tid - RAD) * (tid - RAD)) * (1.0f / 72.0f));

    // ---- stage y halo: 21*26*26 elements -------------------------------
    const float* ybase = yprob + (size_t)n * NCLS * HW * HW;
    for (int i = tid; i < NCLS * HALO * HALO; i += 256) {
        int c  = i / (HALO * HALO);
        int j  = i - c * (HALO * HALO);
        int ly = j / HALO, lx = j - ly * HALO;
        int gy = gy0 + ly, gx = gx0 + lx;
        float* lp = (c < 20) ? &sy4[c >> 2][ly][lx][c & 3] : &sy1[ly][lx];
        if ((unsigned)gy < HW && (unsigned)gx < HW) {
            unsigned long long ga =
                (unsigned long long)(ybase + ((size_t)c * HW + gy) * HW + gx);
            uint32_t la = (uint32_t)(uintptr_t)lp;   // low 32b of flat = LDS addr
            asm volatile("global_load_async_to_lds_b32 %0, %1, off"
                         :: "v"(la), "v"(ga) : "memory");
        } else {
            *lp = 0.0f;                               // zero padding semantics
        }
    }
    // ---- stage rgb_ds halo: 3*26*26 elements ---------------------------
    const float* rbase = rgbds + (size_t)n * 3 * HW * HW;
    for (int i = tid; i < 3 * HALO * HALO; i += 256) {
        int c  = i / (HALO * HALO);
        int j  = i - c * (HALO * HALO);
        int ly = j / HALO, lx = j - ly * HALO;
        int gy = gy0 + ly, gx = gx0 + lx;
        float* lp = &sr[c][ly][lx];
        if ((unsigned)gy < HW && (unsigned)gx < HW) {
            unsigned long long ga =
                (unsigned long long)(rbase + ((size_t)c * HW + gy) * HW + gx);
            uint32_t la = (uint32_t)(uintptr_t)lp;
            asm volatile("global_load_async_to_lds_b32 %0, %1, off"
                         :: "v"(la), "v"(ga) : "memory");
        } else {
            *lp = 0.0f;
        }
    }
    asm volatile("s_wait_asynccnt 0x0" ::: "memory");
    __syncthreads();

    // ---- per-pixel setup ------------------------------------------------
    const int lx = tid & 15, ly = tid >> 4;
    const int gx = tx * TILE + lx;
    const int gy = ty * TILE + ly;

    float4 ycv[5];
#pragma unroll
    for (int b = 0; b < 5; ++b)
        ycv[b] = *(const float4*)&sy4[b][ly + RAD][lx + RAD][0];
    const float yc20 = sy1[ly + RAD][lx + RAD];
    const float r0 = sr[0][ly + RAD][lx + RAD];
    const float r1 = sr[1][ly + RAD][lx + RAD];
    const float r2 = sr[2][ly + RAD][lx + RAD];

    // out-of-image neighbors have all-zero padded features:
    // xy distance collapses to the center norm -> exp(-(gx^2+gy^2)/72)
    const float ecxy = __expf((float)(-(gx * gx + gy * gy)) * (1.0f / 72.0f));

    float acc = 0.0f;
    for (int dy = 0; dy < DIA; ++dy) {
        const int py = gy + dy - RAD;
        const bool rowin = (unsigned)py < HW;
        const float eY = __expf((float)(-(dy - RAD) * (dy - RAD)) * (1.0f / 72.0f));
        const int yy = ly + dy;
#pragma unroll 2
        for (int dx = 0; dx < DIA; ++dx) {
            const int px = gx + dx - RAD;
            const bool inb = rowin && ((unsigned)px < HW);
            const float gX = sgx[dx];        // unconditional load -> cndmask
            float exy = eY * gX;
            if (!inb) exy = ecxy;

            const int xx = lx + dx;
            const float d0 = r0 - sr[0][yy][xx];
            const float d1 = r1 - sr[1][yy][xx];
            const float d2 = r2 - sr[2][yy][xx];
            const float srgb = d0 * d0 + d1 * d1 + d2 * d2;
            const float w = exy * (0.9f * __expf(-0.5f * srgb) + 0.1f);

            float s = 0.0f;
#pragma unroll
            for (int b = 0; b < 5; ++b) {
                const float4 v = *(const float4*)&sy4[b][yy][xx][0];
                s += fabsf(ycv[b].x - v.x) + fabsf(ycv[b].y - v.y) +
                     fabsf(ycv[b].z - v.z) + fabsf(ycv[b].w - v.w);
            }
            s += fabsf(yc20 - sy1[yy][xx]);
            acc += w * s;
        }
    }
    acc *= (1.0f / 688128.0f);   // mean over N*C*H*W = 2*21*128*128

    // ---- deterministic block reduction (WMMA wave sum) ------------------
    float rs = wave_sum_wmma(acc);
    const int wave = tid >> 5, lane = tid & 31;
    if (lane == 0) swsum[wave] = rs;
    __syncthreads();
    if (tid == 0) {
        float t = 0.0f;
#pragma unroll
        for (int wv = 0; wv < 8; ++wv) t += swsum[wv];
        partials[tx + 8 * ty + 64 * n] = t;
    }
}

// ---------------------------------------------------------------------------
// Kernel 3: deterministic final reduction of 128 block partials (1 block,
// 4 waves), fixed dataflow -> bit-identical across graph replays.
// ---------------------------------------------------------------------------
__global__ __launch_bounds__(NBLK) void final_reduce_kernel(
    const float* __restrict__ partials, float* __restrict__ out)
{
    __shared__ float sw[4];
    const int tid = threadIdx.x;
    const float v = partials[tid];
    const float rs = wave_sum_wmma(v);
    if ((tid & 31) == 0) sw[tid >> 5] = rs;
    __syncthreads();
    if (tid == 0)
        out[0] = ((sw[0] + sw[1]) + sw[2]) + sw[3];
}

// ---------------------------------------------------------------------------
extern "C" void kernel_launch(void* const* d_in, const int* in_sizes, int n_in,
                              void* d_out, int out_size, void* d_ws, size_t ws_size,
                              hipStream_t stream) {
    (void)in_sizes; (void)n_in; (void)out_size; (void)ws_size;
    const float* y   = (const float*)d_in[0];   // (2,21,128,128) f32
    const float* rgb = (const float*)d_in[1];   // (2,3,512,512)  f32
    // d_in[2..4] = height_input(512), width_input(512), kernels_radius(5): fixed.
    float* out      = (float*)d_out;
    float* rgbds    = (float*)d_ws;                          // 384 KiB
    float* partials = (float*)d_ws + 2 * 3 * HW * HW;        // 128 floats

    rgb_pool_kernel<<<dim3((2 * 3 * HW * HW + 255) / 256), dim3(256), 0, stream>>>(
        rgb, rgbds);

    crf_loss_kernel<<<dim3(HW / TILE, HW / TILE, 2), dim3(256), 0, stream>>>(
        y, rgbds, partials);

    final_reduce_kernel<<<dim3(1), dim3(NBLK), 0, stream>>>(partials, out);
}